// GATBlock_64304250356385
// MI455X (gfx1250) — compile-verified
//
#include <hip/hip_runtime.h>

// Problem constants (match reference)
#define NN      50000
#define NE      400000
#define EF      (NE + NN)      // 450000 edges incl. self loops
#define IN_DIM  128
#define OUT_DIM 32
#define EDGE_DIM 64
#define HEADS   8
#define HC      256            // HEADS*OUT_DIM
#define NEG_SLOPE 0.2f

typedef __attribute__((ext_vector_type(2))) float v2f;
typedef __attribute__((ext_vector_type(8))) float v8f;

// ---- order-preserving float <-> uint map for atomicMax-based segment max ----
__device__ __forceinline__ unsigned flip_f32(float f) {
  unsigned u = __float_as_uint(f);
  return (u & 0x80000000u) ? ~u : (u | 0x80000000u);
}
__device__ __forceinline__ float unflip_f32(unsigned u) {
  u = (u & 0x80000000u) ? (u & 0x7fffffffu) : ~u;
  return __uint_as_float(u);
}

// ---------------------------------------------------------------- zero fill
__global__ void zero_kernel(float* __restrict__ p, long n) {
  long i = (long)blockIdx.x * blockDim.x + threadIdx.x;
  if (i < n) p[i] = 0.0f;
}

// ------------------------------------------------- degree + edge_attr sums
__global__ void edge_stats_kernel(const int* __restrict__ ei,
                                  const int* __restrict__ rix,
                                  const float* __restrict__ rel,
                                  float* __restrict__ attr_sum,
                                  float* __restrict__ deg) {
  int e = blockIdx.x * blockDim.x + threadIdx.x;
  if (e >= NE) return;
  int d = ei[NE + e];
  int r = rix[e];
  atomicAdd(&deg[d], 1.0f);
  const float* rp = rel + (long)r * EDGE_DIM;
  float* ap = attr_sum + (long)d * EDGE_DIM;
#pragma unroll 4
  for (int c = 0; c < EDGE_DIM; ++c) atomicAdd(&ap[c], rp[c]);
}

// ------------------------------------------- self_attr = attr_sum / max(deg,1)
__global__ void self_attr_kernel(float* __restrict__ attr,
                                 const float* __restrict__ deg) {
  long i = (long)blockIdx.x * blockDim.x + threadIdx.x;
  if (i >= (long)NN * EDGE_DIM) return;
  float dv = deg[i >> 6];
  attr[i] = attr[i] / fmaxf(dv, 1.0f);
}

// --------------------------------- x@W + b projections via f32 WMMA 16x16x4
// One wave computes one 16x16 tile. A (16x4 f32): lanes 0-15 hold M=lane,
// K={k0,k0+1}; lanes 16-31 hold M=lane-16, K={k0+2,k0+3}. B mirrors with N.
__global__ void __launch_bounds__(32)
proj_gemm_kernel(const float* __restrict__ x,
                 const float* __restrict__ Wl, const float* __restrict__ bl,
                 const float* __restrict__ Wr, const float* __restrict__ br,
                 float* __restrict__ xl, float* __restrict__ xr) {
  const int lane  = threadIdx.x;
  const int col0  = blockIdx.x * 16;     // 16 col tiles (HC=256)
  const int row0  = blockIdx.y * 16;     // 3125 row tiles (NN=50000)
  const int which = blockIdx.z;          // 0 -> W_l, 1 -> W_r
  const float* W  = which ? Wr : Wl;
  const float* bv = which ? br : bl;
  float* dst      = which ? xr : xl;

  const int m    = lane & 15;
  const int koff = (lane >> 4) << 1;     // 0 or 2
  const float* arow = x + (long)(row0 + m) * IN_DIM;
  const int col = col0 + m;

  v8f acc = {};
#pragma unroll
  for (int k0 = 0; k0 < IN_DIM; k0 += 4) {
    v2f a = *(const v2f*)(arow + k0 + koff);
    v2f b;
    b.x = W[(long)(k0 + koff) * HC + col];
    b.y = W[(long)(k0 + koff + 1) * HC + col];
    acc = __builtin_amdgcn_wmma_f32_16x16x4_f32(false, a, false, b,
                                                (short)0, acc, false, false);
  }
  // C/D layout: VGPR r, lanes 0-15 -> M=r ; lanes 16-31 -> M=r+8 ; N=lane&15
  const int rbase = row0 + ((lane >> 4) << 3);
  const float bcol = bv[col];
#pragma unroll
  for (int r = 0; r < 8; ++r)
    dst[(long)(rbase + r) * HC + col] = acc[r] + bcol;
}

// ----------- fused: e_emb = ea@W_e (WMMA) + gathers + leakyReLU + dot(att)
// One wave per 16-edge block (NE and EF both divisible by 16, so a block is
// either all regular edges or all self loops).
__global__ void __launch_bounds__(32)
logits_kernel(const float* __restrict__ xl, const float* __restrict__ xr,
              const float* __restrict__ We, const float* __restrict__ att,
              const int* __restrict__ ei, const int* __restrict__ rix,
              const float* __restrict__ rel, const float* __restrict__ self_attr,
              float* __restrict__ logits) {
  __shared__ float sl[16 * HEADS];
  __shared__ int s_src[16], s_dst[16];

  const int lane = threadIdx.x;
  const long e0  = (long)blockIdx.x * 16;
  const int m    = lane & 15;
  const long eg  = e0 + m;

  const float* arow;
  int srcv, dstv;
  if (eg < NE) {
    srcv = ei[eg];
    dstv = ei[NE + eg];
    arow = rel + (long)rix[eg] * EDGE_DIM;
  } else {
    int node = (int)(eg - NE);
    srcv = node; dstv = node;
    arow = self_attr + (long)node * EDGE_DIM;
  }
  if (lane < 16) { s_src[lane] = srcv; s_dst[lane] = dstv; }
#pragma unroll
  for (int i = 0; i < 4; ++i) sl[lane * 4 + i] = 0.0f;
  __syncthreads();

  const int koff  = (lane >> 4) << 1;
  const int nlane = lane & 15;
  const int half8 = (lane >> 4) << 3;

  for (int ct = 0; ct < 16; ++ct) {          // 16 column tiles of HC=256
    const int col = ct * 16 + nlane;
    v8f acc = {};
#pragma unroll
    for (int k0 = 0; k0 < EDGE_DIM; k0 += 4) {
      v2f a = *(const v2f*)(arow + k0 + koff);
      v2f b;
      b.x = We[(long)(k0 + koff) * HC + col];
      b.y = We[(long)(k0 + koff + 1) * HC + col];
      acc = __builtin_amdgcn_wmma_f32_16x16x4_f32(false, a, false, b,
                                                  (short)0, acc, false, false);
    }
    const float av = att[col];               // att flattened over [h][c] == col
    const int h = ct >> 1;                   // 2 col tiles per head (32 cols)
#pragma unroll
    for (int r = 0; r < 8; ++r) {
      const int el = r + half8;              // edge index in C-layout
      float mm = acc[r]
               + xl[(long)s_src[el] * HC + col]
               + xr[(long)s_dst[el] * HC + col];
      mm = (mm >= 0.0f) ? mm : NEG_SLOPE * mm;
      float p = mm * av;
      p += __shfl_xor(p, 1, 16);
      p += __shfl_xor(p, 2, 16);
      p += __shfl_xor(p, 4, 16);
      p += __shfl_xor(p, 8, 16);
      if (nlane == 0) sl[el * HEADS + h] += p;
    }
  }
  __syncthreads();
#pragma unroll
  for (int i = 0; i < 4; ++i) {
    int idx = lane * 4 + i;                  // [16 edges][8 heads] contiguous
    logits[e0 * HEADS + idx] = sl[idx];
  }
}

// ----------------------------------------------- segment max (flip-atomicMax)
__global__ void lmax_kernel(const float* __restrict__ logits,
                            const int* __restrict__ ei,
                            unsigned* __restrict__ lmaxk) {
  long i = (long)blockIdx.x * blockDim.x + threadIdx.x;
  if (i >= (long)EF * HEADS) return;
  long e = i >> 3;
  int h = (int)(i & 7);
  int d = (e < NE) ? ei[NE + e] : (int)(e - NE);
  atomicMax(&lmaxk[(long)d * HEADS + h], flip_f32(logits[i]));
}

// ------------------------------------------- exp(logit - max) + segment sum
__global__ void exp_kernel(float* __restrict__ logits,
                           const int* __restrict__ ei,
                           const unsigned* __restrict__ lmaxk,
                           float* __restrict__ denom) {
  long i = (long)blockIdx.x * blockDim.x + threadIdx.x;
  if (i >= (long)EF * HEADS) return;
  long e = i >> 3;
  int h = (int)(i & 7);
  int d = (e < NE) ? ei[NE + e] : (int)(e - NE);
  float ex = expf(logits[i] - unflip_f32(lmaxk[(long)d * HEADS + h]));
  logits[i] = ex;                            // overwrite logits with exp
  atomicAdd(&denom[(long)d * HEADS + h], ex);
}

// ----------------------------------- scatter: out_hc[dst] += alpha * xl[src]
__global__ void agg_kernel(const float* __restrict__ ex,
                           const float* __restrict__ denom,
                           const float* __restrict__ xl,
                           const int* __restrict__ ei,
                           float* __restrict__ outhc) {
  long i = (long)blockIdx.x * blockDim.x + threadIdx.x;
  if (i >= (long)EF * HC) return;
  long e = i >> 8;
  int c = (int)(i & 255);
  int s, d;
  if (e < NE) { s = ei[e]; d = ei[NE + e]; }
  else        { s = (int)(e - NE); d = s; }
  int h = c >> 5;
  float alpha = ex[e * HEADS + h] / denom[(long)d * HEADS + h];
  atomicAdd(&outhc[(long)d * HC + c], alpha * xl[(long)s * HC + c]);
}

// ---------------------------------------------------- head mean + bias -> out
__global__ void finalize_kernel(const float* __restrict__ outhc,
                                const float* __restrict__ bias,
                                float* __restrict__ out) {
  long i = (long)blockIdx.x * blockDim.x + threadIdx.x;
  if (i >= (long)NN * OUT_DIM) return;
  long n = i >> 5;
  int c = (int)(i & 31);
  float s = 0.0f;
#pragma unroll
  for (int h = 0; h < HEADS; ++h) s += outhc[n * HC + h * OUT_DIM + c];
  out[i] = s * 0.125f + bias[c];
}

// ------------------------------------------------- passthrough for relations
__global__ void copy_kernel(const float* __restrict__ src,
                            float* __restrict__ dst, int n) {
  int i = blockIdx.x * blockDim.x + threadIdx.x;
  if (i < n) dst[i] = src[i];
}

extern "C" void kernel_launch(void* const* d_in, const int* in_sizes, int n_in,
                              void* d_out, int out_size, void* d_ws, size_t ws_size,
                              hipStream_t stream) {
  const float* x         = (const float*)d_in[0];
  const int*   ei        = (const int*)d_in[1];
  const float* relations = (const float*)d_in[2];
  const int*   rix       = (const int*)d_in[3];
  const float* W_l       = (const float*)d_in[4];
  const float* b_l       = (const float*)d_in[5];
  const float* W_r       = (const float*)d_in[6];
  const float* b_r       = (const float*)d_in[7];
  const float* W_e       = (const float*)d_in[8];
  const float* att       = (const float*)d_in[9];
  const float* bias      = (const float*)d_in[10];
  float* out = (float*)d_out;
  char*  ws  = (char*)d_ws;

  // Workspace layout (bytes)
  const size_t XL_OFF    = 0;                               // NN*HC*4
  const size_t XR_OFF    = XL_OFF + (size_t)NN * HC * 4;
  const size_t ATTR_OFF  = XR_OFF + (size_t)NN * HC * 4;    // NN*EDGE_DIM*4
  const size_t DEG_OFF   = ATTR_OFF + (size_t)NN * EDGE_DIM * 4;
  const size_t LOG_OFF   = DEG_OFF + (size_t)NN * 4;        // EF*HEADS*4
  const size_t LMAX_OFF  = LOG_OFF + (size_t)EF * HEADS * 4;
  const size_t DEN_OFF   = LMAX_OFF + (size_t)NN * HEADS * 4;
  const size_t OUTHC_OFF = DEN_OFF + (size_t)NN * HEADS * 4;

  float*    xl    = (float*)(ws + XL_OFF);
  float*    xr    = (float*)(ws + XR_OFF);
  float*    attr  = (float*)(ws + ATTR_OFF);
  float*    deg   = (float*)(ws + DEG_OFF);
  float*    lg    = (float*)(ws + LOG_OFF);
  unsigned* lmaxk = (unsigned*)(ws + LMAX_OFF);
  float*    den   = (float*)(ws + DEN_OFF);
  float*    outhc = (float*)(ws + OUTHC_OFF);

  // Zero accumulators: attr+deg (contiguous), lmax+den (contiguous; 0u == key
  // below any finite logit), outhc.
  {
    long n1 = (long)NN * EDGE_DIM + NN;
    zero_kernel<<<(n1 + 255) / 256, 256, 0, stream>>>(attr, n1);
    long n2 = (long)NN * HEADS * 2;
    zero_kernel<<<(n2 + 255) / 256, 256, 0, stream>>>((float*)lmaxk, n2);
    long n3 = (long)NN * HC;
    zero_kernel<<<(n3 + 255) / 256, 256, 0, stream>>>(outhc, n3);
  }

  // 1) degree + edge_attr segment sums over dst
  edge_stats_kernel<<<(NE + 255) / 256, 256, 0, stream>>>(ei, rix, relations, attr, deg);

  // 2) self-loop attr = mean of incoming edge_attr (in place)
  {
    long n = (long)NN * EDGE_DIM;
    self_attr_kernel<<<(n + 255) / 256, 256, 0, stream>>>(attr, deg);
  }

  // 3) x_l = x@W_l+b_l, x_r = x@W_r+b_r  (fp32 WMMA, 16x16 tiles)
  proj_gemm_kernel<<<dim3(HC / 16, NN / 16, 2), 32, 0, stream>>>(
      x, W_l, b_l, W_r, b_r, xl, xr);

  // 4) fused e_emb GEMM + GATv2 logits (never materializes e_emb)
  logits_kernel<<<EF / 16, 32, 0, stream>>>(
      xl, xr, W_e, att, ei, rix, relations, attr, lg);

  // 5-6) segment softmax over dst
  {
    long n = (long)EF * HEADS;
    lmax_kernel<<<(n + 255) / 256, 256, 0, stream>>>(lg, ei, lmaxk);
    exp_kernel<<<(n + 255) / 256, 256, 0, stream>>>(lg, ei, lmaxk, den);
  }

  // 7) scatter-add aggregation of alpha-weighted x_l[src]
  {
    long n = (long)EF * HC;
    agg_kernel<<<(n + 255) / 256, 256, 0, stream>>>(lg, den, xl, ei, outhc);
  }

  // 8) head mean + bias
  {
    long n = (long)NN * OUT_DIM;
    finalize_kernel<<<(n + 255) / 256, 256, 0, stream>>>(outhc, bias, out);
  }

  // 9) second tuple element: relations passthrough
  {
    int n = 512 * 64;
    copy_kernel<<<(n + 255) / 256, 256, 0, stream>>>(relations, out + (long)NN * OUT_DIM, n);
  }
}